// CPCircuitLayer_52278341927190
// MI455X (gfx1250) — compile-verified
//
#include <hip/hip_runtime.h>
#include <hip/hip_bf16.h>

typedef float v2f __attribute__((ext_vector_type(2)));
typedef float v8f __attribute__((ext_vector_type(8)));

namespace {
constexpr int kS  = 256;   // sequence length
constexpr int kH  = 2048;  // hidden size
constexpr int kR  = 64;    // rank
constexpr int kNH = 16;    // heads
}

// -----------------------------------------------------------------------------
// Stage 1: E[s][r] = sum_h hs[s][h] * W[r][h]   (E1 from W1, E2 from W2)
// One wave per (16-row tile of S, which-W). Each wave computes a 16x64 strip,
// reusing the A fragment across 4 N-tiles per WMMA K-step (K=4, fp32 WMMA).
// fp32 WMMA fragment layout (16x16x4):
//   A: lane L -> m = L&15, k = 2*(L>>4) + {0,1}  (VGPR0, VGPR1)
//   B: lane L -> n = L&15, k = 2*(L>>4) + {0,1}
//   C/D: lane L -> n = L&15, M = v + 8*(L>>4) for VGPR v
// -----------------------------------------------------------------------------
__global__ __launch_bounds__(128) void cp_proj_kernel(
    const float* __restrict__ hs,
    const float* __restrict__ W1,
    const float* __restrict__ W2,
    float* __restrict__ E1,
    float* __restrict__ E2)
{
    const int lane  = threadIdx.x & 31;
    const int wave  = threadIdx.x >> 5;
    const int w     = blockIdx.x * 4 + wave;   // 0..31
    const int which = w >> 4;                  // 0 -> W1/E1, 1 -> W2/E2
    const int tileM = w & 15;                  // 16 row tiles of S

    const float* __restrict__ W = which ? W2 : W1;
    float* __restrict__       E = which ? E2 : E1;

    const int mn   = lane & 15;        // m for A, n for B/D
    const int half = lane >> 4;
    const int kb   = half << 1;        // 0 or 2

    const float* pA = hs + (tileM * 16 + mn) * kH + kb;
    const float* pB = W  + mn * kH + kb;   // N-tile t adds t*16 rows of W

    v8f acc0 = {}, acc1 = {}, acc2 = {}, acc3 = {};

    for (int k0 = 0; k0 < kH; k0 += 4) {
        v2f a  = *(const v2f*)(pA + k0);
        __builtin_prefetch(pA + k0 + 64, 0, 3);   // global_prefetch ahead on A stream
        v2f b0 = *(const v2f*)(pB + 0 * 16 * kH + k0);
        v2f b1 = *(const v2f*)(pB + 1 * 16 * kH + k0);
        v2f b2 = *(const v2f*)(pB + 2 * 16 * kH + k0);
        v2f b3 = *(const v2f*)(pB + 3 * 16 * kH + k0);
        acc0 = __builtin_amdgcn_wmma_f32_16x16x4_f32(false, a, false, b0, (short)0, acc0, false, false);
        acc1 = __builtin_amdgcn_wmma_f32_16x16x4_f32(false, a, false, b1, (short)0, acc1, false, false);
        acc2 = __builtin_amdgcn_wmma_f32_16x16x4_f32(false, a, false, b2, (short)0, acc2, false, false);
        acc3 = __builtin_amdgcn_wmma_f32_16x16x4_f32(false, a, false, b3, (short)0, acc3, false, false);
    }

    // D layout: VGPR v holds row M = v + 8*half, column n = mn.
    float* pE = E + (tileM * 16 + half * 8) * kR + mn;
    #pragma unroll
    for (int v = 0; v < 8; ++v) {
        pE[v * kR +  0] = acc0[v];
        pE[v * kR + 16] = acc1[v];
        pE[v * kR + 32] = acc2[v];
        pE[v * kR + 48] = acc3[v];
    }
}

// -----------------------------------------------------------------------------
// Stage 2: out[h,i,j] = sum_r (E1[i,r]*hm[h,r]*cp[r]) * E2[j,r]
// One wave per 16x16 output tile: NH * (S/16) * (S/16) = 4096 tiles.
// K = 64 -> 16 fp32 WMMA steps, fully unrolled.
// -----------------------------------------------------------------------------
__global__ __launch_bounds__(256) void cp_outer_kernel(
    const float* __restrict__ E1,
    const float* __restrict__ E2,
    const float* __restrict__ hm,
    const float* __restrict__ cp,
    float* __restrict__ out)
{
    const int lane = threadIdx.x & 31;
    const int w    = blockIdx.x * 8 + (threadIdx.x >> 5);  // 0..4095
    const int tj   = w & 15;
    const int ti   = (w >> 4) & 15;
    const int h    = w >> 8;

    const int mn   = lane & 15;
    const int half = lane >> 4;
    const int kb   = half << 1;

    const float* pA = E1 + (ti * 16 + mn) * kR + kb;
    const float* pB = E2 + (tj * 16 + mn) * kR + kb;
    const float* pS = hm + h * kR + kb;
    const float* pC = cp + kb;

    v8f acc = {};
    #pragma unroll
    for (int k0 = 0; k0 < kR; k0 += 4) {
        v2f sh = *(const v2f*)(pS + k0);
        v2f sc = *(const v2f*)(pC + k0);
        v2f av = *(const v2f*)(pA + k0);
        v2f bv = *(const v2f*)(pB + k0);
        v2f a;
        a.x = av.x * sh.x * sc.x;
        a.y = av.y * sh.y * sc.y;
        acc = __builtin_amdgcn_wmma_f32_16x16x4_f32(false, a, false, bv, (short)0, acc, false, false);
    }

    // out[h][i][j], i = ti*16 + v + 8*half, j = tj*16 + mn
    float* pO = out + (h * kS + ti * 16 + half * 8) * kS + tj * 16 + mn;
    #pragma unroll
    for (int v = 0; v < 8; ++v)
        pO[v * kS] = acc[v];
}

extern "C" void kernel_launch(void* const* d_in, const int* in_sizes, int n_in,
                              void* d_out, int out_size, void* d_ws, size_t ws_size,
                              hipStream_t stream) {
    (void)in_sizes; (void)n_in; (void)out_size; (void)ws_size;

    const float* hs = (const float*)d_in[0];
    // d_in[1] = all_indices: full (h,i,j) meshgrid in h-major order -> identity
    // re-indexing of the factorized result; not needed.
    const float* W1 = (const float*)d_in[2];
    const float* W2 = (const float*)d_in[3];
    const float* hm = (const float*)d_in[4];
    const float* cp = (const float*)d_in[5];
    float* out = (float*)d_out;

    float* E1 = (float*)d_ws;        // 256*64 floats = 64 KB
    float* E2 = E1 + kS * kR;        // next 64 KB

    hipLaunchKernelGGL(cp_proj_kernel, dim3(8), dim3(128), 0, stream,
                       hs, W1, W2, E1, E2);
    hipLaunchKernelGGL(cp_outer_kernel, dim3(512), dim3(256), 0, stream,
                       E1, E2, hm, cp, out);
}